// CorrelationLayer_51324859187627
// MI455X (gfx1250) — compile-verified
//
#include <hip/hip_runtime.h>
#include <hip/hip_bf16.h>

// ---------------------------------------------------------------------------
// Correlation layer (banded Gram matrix) on MI455X / gfx1250.
//   out[b, (dy+4)*9+(dx+4), h, w] = <f1n[b,:,h,w], f2n[b,:,h+dy,w+dx]>
// with f1n/f2n L2-normalized along C and zero padding on f2.
//
// Strategy:
//   Pass 1: inverse L2 norms of f1/f2 pixels into workspace.
//   Pass 2: per (b,h,64-pixel span) block: each of 4 waves owns a 16-pixel
//     A tile (fp32, scaled by invn1) held in 64 VGPRs; for each dy it runs
//     V_WMMA_F32_16X16X4_F32 against two 16-column f2 tiles covering the
//     24-pixel band, scales by invn2, stages the 9x64 (dx x pixel) slab in
//     LDS, and the block stores it coalesced (64-contiguous per plane).
// ---------------------------------------------------------------------------

#define PD   4
#define NSH  81
#define Cc   128
#define Hh   184
#define Ww   320
#define Bb   4
#define HWc  (Hh * Ww)          // 58880

typedef __attribute__((ext_vector_type(2))) float v2f;
typedef __attribute__((ext_vector_type(8))) float v8f;

__device__ __forceinline__ int iclamp(int v, int lo, int hi) {
    return v < lo ? lo : (v > hi ? hi : v);
}

// ---------------------------------------------------------------------------
// Pass 1: invn[idx] = 1 / max(||feat[b,:,h,w]||_2, 1e-12)
//   idx in [0, B*HW)       -> feat1
//   idx in [B*HW, 2*B*HW)  -> feat2
// ---------------------------------------------------------------------------
__global__ void inv_norm_kernel(const float* __restrict__ f1,
                                const float* __restrict__ f2,
                                float* __restrict__ invn) {
    const int total = Bb * HWc;
    int idx = blockIdx.x * blockDim.x + threadIdx.x;
    if (idx >= 2 * total) return;
    const float* src = (idx < total) ? f1 : f2;
    int pix = (idx < total) ? idx : idx - total;
    int b   = pix / HWc;
    int hw  = pix - b * HWc;
    const float* p = src + (size_t)b * Cc * HWc + hw;
    float s = 0.0f;
#pragma unroll 8
    for (int k = 0; k < Cc; ++k) {
        float v = p[(size_t)k * HWc];
        s += v * v;
    }
    invn[idx] = 1.0f / fmaxf(sqrtf(s), 1e-12f);
}

// ---------------------------------------------------------------------------
// Pass 2: banded-Gram WMMA correlation.
// Block: 128 threads = 4 waves. Wave w owns pixels [w0+16*w, w0+16*w+16).
// ---------------------------------------------------------------------------
__global__ __launch_bounds__(128)
void corr_wmma_kernel(const float* __restrict__ f1,
                      const float* __restrict__ f2,
                      const float* __restrict__ invn1,
                      const float* __restrict__ invn2,
                      float* __restrict__ out) {
    __shared__ float stage[9 * 65];          // [dx+4][pixel], stride 65 (bank-pad)

    const int lane = threadIdx.x & 31;
    const int wid  = threadIdx.x >> 5;       // wave id 0..3
    const int half = lane >> 4;              // 0 / 1 (K sub-pair selector)
    const int ln   = lane & 15;              // M (for A) or N (for B/C/D)

    const int w0 = blockIdx.x * 64;          // block pixel base (W multiple of 64? 320=5*64)
    const int h  = blockIdx.y;
    const int b  = blockIdx.z;
    const int t  = w0 + wid * 16;            // wave's A-tile pixel base

    const size_t fbase = (size_t)b * Cc * HWc;
    const size_t nbase = (size_t)b * HWc;

    // ---- Load + pre-scale A tile: a[s] holds K = 4s + 2*half + {0,1}, M = ln
    const int   mp = t + ln;                               // f1 pixel of this lane
    const float s1 = invn1[nbase + (size_t)h * Ww + mp];
    v2f a[32];
    {
        const float* pa = f1 + fbase + (size_t)(half * 2) * HWc + (size_t)h * Ww + mp;
#pragma unroll
        for (int s = 0; s < 32; ++s) {
            a[s].x = pa[0]   * s1;
            a[s].y = pa[HWc] * s1;
            pa += 4 * HWc;
        }
    }

    // ---- Per-lane B columns for the two band tiles (fixed across dy)
    const int  qa   = (t - 4)  + ln;                       // tile-0 column (f2 pixel)
    const int  qb   = (t + 12) + ln;                       // tile-1 column
    const int  qac  = iclamp(qa, 0, Ww - 1);
    const int  qbc  = iclamp(qb, 0, Ww - 1);
    const bool qaok = (qa >= 0) && (qa < Ww);
    const bool qbok = (qb >= 0) && (qb < Ww);

    for (int dy = -PD; dy <= PD; ++dy) {
        const int  row   = h + dy;
        const bool rowok = (row >= 0) && (row < Hh);
        const int  rowc  = iclamp(row, 0, Hh - 1);

        // invn2 doubles as the zero-padding mask (0 when q or row OOB)
        const float i2a = (rowok && qaok) ? invn2[nbase + (size_t)rowc * Ww + qac] : 0.0f;
        const float i2b = (rowok && qbok) ? invn2[nbase + (size_t)rowc * Ww + qbc] : 0.0f;

        v8f acc0 = {0.f,0.f,0.f,0.f,0.f,0.f,0.f,0.f};
        v8f acc1 = {0.f,0.f,0.f,0.f,0.f,0.f,0.f,0.f};

        const float* pba = f2 + fbase + (size_t)(half * 2) * HWc + (size_t)rowc * Ww + qac;
        const float* pbb = f2 + fbase + (size_t)(half * 2) * HWc + (size_t)rowc * Ww + qbc;
#pragma unroll
        for (int s = 0; s < 32; ++s) {
            v2f bva; bva.x = pba[0]; bva.y = pba[HWc];
            v2f bvb; bvb.x = pbb[0]; bvb.y = pbb[HWc];
            // D = A(16x4) * B(4x16) + C, exact fp32
            acc0 = __builtin_amdgcn_wmma_f32_16x16x4_f32(
                       false, a[s], false, bva, (short)0, acc0, false, false);
            acc1 = __builtin_amdgcn_wmma_f32_16x16x4_f32(
                       false, a[s], false, bvb, (short)0, acc1, false, false);
            pba += 4 * HWc;
            pbb += 4 * HWc;
        }

        // ---- Scale by invn2 and stage useful band into LDS [dx+4][local pixel]
#pragma unroll
        for (int v = 0; v < 8; ++v) {
            const int m  = v + 8 * half;      // C/D row: M = v + 8*(lane>=16)
            const int pl = wid * 16 + m;      // pixel local to block (0..63)
            const int dxa = qa - (t + m);
            if (dxa >= -PD && dxa <= PD)
                stage[(dxa + PD) * 65 + pl] = acc0[v] * i2a;
            const int dxb = qb - (t + m);
            if (dxb >= -PD && dxb <= PD)
                stage[(dxb + PD) * 65 + pl] = acc1[v] * i2b;
        }
        __syncthreads();

        // ---- Coalesced store: 9 planes x 64 contiguous pixels
        for (int e = threadIdx.x; e < 9 * 64; e += 128) {
            const int dxi  = e >> 6;
            const int pl   = e & 63;
            const int sidx = (dy + PD) * 9 + dxi;
            out[(((size_t)b * NSH + sidx) * Hh + h) * Ww + w0 + pl] =
                stage[dxi * 65 + pl];
        }
        __syncthreads();
    }
}

// ---------------------------------------------------------------------------
extern "C" void kernel_launch(void* const* d_in, const int* in_sizes, int n_in,
                              void* d_out, int out_size, void* d_ws, size_t ws_size,
                              hipStream_t stream) {
    (void)in_sizes; (void)n_in; (void)out_size; (void)ws_size;
    const float* f1 = (const float*)d_in[0];
    const float* f2 = (const float*)d_in[1];
    float* invn = (float*)d_ws;              // 2 * B*HW floats (~1.9 MB)
    float* outp = (float*)d_out;

    const int total = Bb * HWc;
    {
        int threads = 256;
        int blocks  = (2 * total + threads - 1) / threads;
        hipLaunchKernelGGL(inv_norm_kernel, dim3(blocks), dim3(threads), 0, stream,
                           f1, f2, invn);
    }
    {
        dim3 grid(Ww / 64, Hh, Bb);          // 5 x 184 x 4 blocks
        hipLaunchKernelGGL(corr_wmma_kernel, grid, dim3(128), 0, stream,
                           f1, f2, invn, invn + total, outp);
    }
}